// GCN_27324581937505
// MI455X (gfx1250) — compile-verified
//
#include <hip/hip_runtime.h>

typedef __attribute__((ext_vector_type(2))) float v2f;
typedef __attribute__((ext_vector_type(8))) float v8f;

#define GCN_N     100000
#define GCN_E     1600000
#define GCN_D     128
#define GCN_SLABS (GCN_N / 16)   // 6250 slabs of 16 rows
#define LDT       132            // padded transposed-weight row stride (dwords)

// ---------------------------------------------------------------------------
// GEMM: OUT[slab rows] = (RELU_IN ? relu(X) : X) @ W  (+ bias if ADD_BIAS)
// W is 128x128 row-major (k-major). One 16-row slab per wave, f32 WMMA 16x16x4.
// ---------------------------------------------------------------------------
template<bool ADD_BIAS, bool RELU_IN>
__global__ __launch_bounds__(256)
void gemm_wmma_f32(const float* __restrict__ X, const float* __restrict__ W,
                   const float* __restrict__ bias, float* __restrict__ OUT)
{
  extern __shared__ float wt[];  // transposed W: wt[n*LDT + k], 128 x LDT floats

  // Cooperative transpose of W into LDS (bank-conflict-free layout for B loads)
  for (int i = threadIdx.x; i < GCN_D * GCN_D; i += 256) {
    int k = i >> 7, n = i & 127;
    wt[n * LDT + k] = W[i];
  }
  __syncthreads();

  const int lane = threadIdx.x & 31;
  const int mrow = lane & 15;           // row within slab (A) / column within tile (B,C)
  const int ksel = (lane >> 4) << 1;    // 0 for lanes 0-15, 2 for lanes 16-31
  const int wavesTotal = (gridDim.x * 256) >> 5;
  const int wave = (blockIdx.x * 256 + threadIdx.x) >> 5;

  for (int slab = wave; slab < GCN_SLABS; slab += wavesTotal) {
    const int row0 = slab << 4;

    // A fragments: lane holds x[row0+mrow][4s + ksel + {0,1}] per k-step s
    const float* xrow = X + (size_t)(row0 + mrow) * GCN_D + ksel;
    v2f a[32];
#pragma unroll
    for (int s = 0; s < 32; ++s) {
      v2f t = *(const v2f*)(xrow + (s << 2));
      if (RELU_IN) { t.x = fmaxf(t.x, 0.0f); t.y = fmaxf(t.y, 0.0f); }
      a[s] = t;
    }

#pragma unroll
    for (int t = 0; t < 8; ++t) {                 // 8 N-tiles of 16 columns
      const float* wp = wt + (t * 16 + mrow) * LDT + ksel;
      v8f acc = {};
#pragma unroll
      for (int s = 0; s < 32; ++s) {              // K = 128 in steps of 4
        v2f b = *(const v2f*)(wp + (s << 2));     // ds_load_b64: W[k..k+1][n]
        acc = __builtin_amdgcn_wmma_f32_16x16x4_f32(
            false, a[s], false, b, (short)0, acc, false, false);
      }
      if (ADD_BIAS) {
        float bv = bias[t * 16 + mrow];
#pragma unroll
        for (int j = 0; j < 8; ++j) acc[j] += bv;
      }
      // C/D layout: VGPR j -> row j (lanes 0-15) / row j+8 (lanes 16-31)
      float* op = OUT + (size_t)(row0 + (lane >> 4) * 8) * GCN_D + t * 16 + mrow;
#pragma unroll
      for (int j = 0; j < 8; ++j) op[(size_t)j * GCN_D] = acc[j];
    }
  }
}

// ---------------------------------------------------------------------------
// Edge scatter: OUT[dst[e]] += R[src[e]]  (one edge per wave; 4 f32 atomics/lane)
// ---------------------------------------------------------------------------
__global__ __launch_bounds__(256)
void scatter_rows(const float* __restrict__ R, const int* __restrict__ src,
                  const int* __restrict__ dst, float* __restrict__ OUT)
{
  const int lane = threadIdx.x & 31;
  const int wave = (blockIdx.x * 256 + threadIdx.x) >> 5;
  const int wavesTotal = (gridDim.x * 256) >> 5;
  for (int e = wave; e < GCN_E; e += wavesTotal) {
    const int s = src[e];
    const int d = dst[e];
    const float4 v = *(const float4*)(R + (size_t)s * GCN_D + (lane << 2));
    float* o = OUT + (size_t)d * GCN_D + (lane << 2);
    unsafeAtomicAdd(o + 0, v.x);
    unsafeAtomicAdd(o + 1, v.y);
    unsafeAtomicAdd(o + 2, v.z);
    unsafeAtomicAdd(o + 3, v.w);
  }
}

// ---------------------------------------------------------------------------
// Final in-place ReLU over the output
// ---------------------------------------------------------------------------
__global__ __launch_bounds__(256)
void relu_inplace(float* __restrict__ p, int n4)
{
  int i = blockIdx.x * 256 + threadIdx.x;
  if (i < n4) {
    float4 v = ((const float4*)p)[i];
    v.x = fmaxf(v.x, 0.0f); v.y = fmaxf(v.y, 0.0f);
    v.z = fmaxf(v.z, 0.0f); v.w = fmaxf(v.w, 0.0f);
    ((float4*)p)[i] = v;
  }
}

// ---------------------------------------------------------------------------
extern "C" void kernel_launch(void* const* d_in, const int* in_sizes, int n_in,
                              void* d_out, int out_size, void* d_ws, size_t ws_size,
                              hipStream_t stream)
{
  const float* x   = (const float*)d_in[0];
  const int*   ei  = (const int*)d_in[1];   // edge_index [2, E] flat
  const float* Wr1 = (const float*)d_in[2];
  const float* Wl1 = (const float*)d_in[3];
  const float* b1  = (const float*)d_in[4];
  const float* Wr2 = (const float*)d_in[5];
  const float* Wl2 = (const float*)d_in[6];
  const float* b2  = (const float*)d_in[7];

  float* out = (float*)d_out;
  float* h1  = (float*)d_ws;                       // layer-1 pre-activation (51.2 MB)
  float* r   = h1 + (size_t)GCN_N * GCN_D;         // x @ W_rel scratch     (51.2 MB)

  const int* src = ei;
  const int* dst = ei + GCN_E;

  const size_t ldsB = (size_t)GCN_D * LDT * sizeof(float);  // 67,584 B (< 320 KB/WG)
  const int GB = 416;    // 3328 waves over 6250 slabs
  const int SB = 3200;   // scatter waves

  // Layer 1: h1 = x@W_root1 + b1 ; r = x@W_rel1 ; h1[dst] += r[src]
  gemm_wmma_f32<true,  false><<<GB, 256, ldsB, stream>>>(x, Wr1, b1, h1);
  gemm_wmma_f32<false, false><<<GB, 256, ldsB, stream>>>(x, Wl1, nullptr, r);
  scatter_rows<<<SB, 256, 0, stream>>>(r, src, dst, h1);

  // Layer 2 (ReLU fused into A-load): out = relu(h1)@W_root2 + b2 ; r = relu(h1)@W_rel2
  gemm_wmma_f32<true,  true ><<<GB, 256, ldsB, stream>>>(h1, Wr2, b2, out);
  gemm_wmma_f32<false, true ><<<GB, 256, ldsB, stream>>>(h1, Wl2, nullptr, r);
  scatter_rows<<<SB, 256, 0, stream>>>(r, src, dst, out);

  relu_inplace<<<(GCN_N * GCN_D / 4 + 255) / 256, 256, 0, stream>>>(out, GCN_N * GCN_D / 4);
}